// My_maxpool1_609885356903
// MI455X (gfx1250) — compile-verified
//
#include <hip/hip_runtime.h>
#include <stdint.h>

// ---------------------------------------------------------------------------
// Problem constants (from reference): N=2,000,000 neighbor rows (5 floats),
// T=1,500,000 traindata rows (4 floats), n=1,048,576 selected, G=n/4 groups.
// Output = G x 4 floats (rows gathered from traindata).
// ---------------------------------------------------------------------------

#define WAVE       32          // gfx1250 is wave32-only
#define TILE       1024        // elements per radix tile (32 lanes x 32 items)
#define ITEMS      (TILE / WAVE)
#define RADIX      256
#define NPASS      4
#define SCAN_T     1024
#define NEG_INIT   (-100000.0f)

// Generic pointers to LDS on amdgcn carry the wave-relative LDS byte offset in
// their low 32 bits (aperture check discards the upper bits) — so truncation
// yields exactly the VGPR value async-to-LDS / DS instructions expect.
__device__ __forceinline__ uint32_t lds_off(const void* p) {
  return (uint32_t)(uintptr_t)p;
}

// CDNA5 async global->LDS copy (tracked on ASYNCcnt), per CDNA5 ISA.
// vdst = LDS byte address (VGPR), vaddr = 64-bit global address (VGPR pair).
__device__ __forceinline__ void async_ld_b128(uint32_t ldsoff, const void* gaddr) {
  asm volatile("global_load_async_to_lds_b128 %0, %1, off"
               :: "v"(ldsoff), "v"(gaddr) : "memory");
}
__device__ __forceinline__ void async_wait0() {
  asm volatile("s_wait_asynccnt 0" ::: "memory");
}

// ---------------------------------------------------------------------------
// Build sort keys: ascending order of key == stable descending order of bt,
// matching jnp.argsort(-bt) (stable).
// ---------------------------------------------------------------------------
__global__ void k_init(const float* __restrict__ neighbor,
                       uint32_t* __restrict__ keys,
                       uint32_t* __restrict__ idx, int N) {
  int i = blockIdx.x * blockDim.x + threadIdx.x;
  if (i >= N) return;
  uint32_t u   = __float_as_uint(neighbor[(size_t)i * 5 + 1]);
  uint32_t asc = u ^ ((uint32_t)((int32_t)u >> 31) | 0x80000000u); // ascending-float transform
  keys[i] = ~asc;                                                  // invert -> descending
  idx[i]  = (uint32_t)i;
}

// ---------------------------------------------------------------------------
// Per-tile digit histogram. One wave per tile; tile staged via async-to-LDS.
// hist layout: hist[d * nblk + blk]  (digit-major => linear scan is stable order)
// ---------------------------------------------------------------------------
__global__ __launch_bounds__(WAVE)
void k_hist(const uint32_t* __restrict__ keysIn, uint32_t* __restrict__ hist,
            int nblk, int nelem, int shift) {
  __shared__ __align__(16) uint32_t tK[TILE];
  __shared__ uint32_t cnt[RADIX];
  const int lane = threadIdx.x;
  const int blk  = blockIdx.x;
  const uint32_t base = (uint32_t)blk * TILE;

  const uint32_t lb = lds_off(tK);
  const uint32_t* g = keysIn + base;
#pragma unroll
  for (int k = 0; k < TILE / (WAVE * 4); ++k) {          // 8 x b128 per lane-group
    int c = k * WAVE + lane;
    async_ld_b128(lb + (uint32_t)c * 16u, g + c * 4);
  }
#pragma unroll
  for (int j = 0; j < RADIX / WAVE; ++j) cnt[lane + j * WAVE] = 0u;
  async_wait0();

  for (int item = 0; item < ITEMS; ++item) {
    int e = item * WAVE + lane;
    bool valid = (base + (uint32_t)e) < (uint32_t)nelem;
    uint32_t d = (tK[e] >> shift) & (RADIX - 1);
    uint32_t active = (uint32_t)__ballot(valid);
    uint32_t peers = 0xFFFFFFFFu;
#pragma unroll
    for (int b = 0; b < 8; ++b) {                         // wave32 digit match
      uint32_t bit = (d >> b) & 1u;
      uint32_t bal = (uint32_t)__ballot((int)bit);
      peers &= bit ? bal : ~bal;
    }
    peers &= active;
    if (valid) {
      int leader = __builtin_ctz(peers);
      if (lane == leader) atomicAdd(&cnt[d], (uint32_t)__popc(peers));
    }
  }
#pragma unroll
  for (int j = 0; j < RADIX / WAVE; ++j) {
    int d = lane + j * WAVE;
    hist[(uint32_t)d * (uint32_t)nblk + (uint32_t)blk] = cnt[d];
  }
}

// ---------------------------------------------------------------------------
// Single-block exclusive scan over M = RADIX*nblk counters (digit-major).
// ---------------------------------------------------------------------------
__global__ __launch_bounds__(SCAN_T)
void k_scan(uint32_t* __restrict__ a, int M) {
  __shared__ uint32_t s[SCAN_T];
  int t = threadIdx.x;
  int chunk = (M + SCAN_T - 1) / SCAN_T;
  int beg = t * chunk;
  int end = beg + chunk; if (end > M) end = M;
  uint32_t sum = 0;
  for (int i = beg; i < end; ++i) sum += a[i];
  s[t] = sum;
  __syncthreads();
  for (int off = 1; off < SCAN_T; off <<= 1) {            // Hillis-Steele inclusive
    uint32_t v = (t >= off) ? s[t - off] : 0u;
    __syncthreads();
    s[t] += v;
    __syncthreads();
  }
  uint32_t run = (t == 0) ? 0u : s[t - 1];
  for (int i = beg; i < end; ++i) { uint32_t old = a[i]; a[i] = run; run += old; }
}

// ---------------------------------------------------------------------------
// Stable scatter. LDS digit counters preloaded with scanned global offsets;
// item-major iteration + lane-ordered ranks give LSD stability.
// ---------------------------------------------------------------------------
__global__ __launch_bounds__(WAVE)
void k_scatter(const uint32_t* __restrict__ keysIn, const uint32_t* __restrict__ idxIn,
               uint32_t* __restrict__ keysOut, uint32_t* __restrict__ idxOut,
               const uint32_t* __restrict__ hist, int nblk, int nelem, int shift) {
  __shared__ __align__(16) uint32_t tK[TILE];
  __shared__ __align__(16) uint32_t tI[TILE];
  __shared__ uint32_t offs[RADIX];
  const int lane = threadIdx.x;
  const int blk  = blockIdx.x;
  const uint32_t base = (uint32_t)blk * TILE;

  const uint32_t lbK = lds_off(tK);
  const uint32_t lbI = lds_off(tI);
#pragma unroll
  for (int k = 0; k < TILE / (WAVE * 4); ++k) {
    int c = k * WAVE + lane;
    async_ld_b128(lbK + (uint32_t)c * 16u, keysIn + base + c * 4);
    async_ld_b128(lbI + (uint32_t)c * 16u, idxIn  + base + c * 4);
  }
#pragma unroll
  for (int j = 0; j < RADIX / WAVE; ++j) {
    int d = lane + j * WAVE;
    offs[d] = hist[(uint32_t)d * (uint32_t)nblk + (uint32_t)blk];
  }
  async_wait0();

  for (int item = 0; item < ITEMS; ++item) {
    int e = item * WAVE + lane;
    bool valid = (base + (uint32_t)e) < (uint32_t)nelem;
    uint32_t key = tK[e];
    uint32_t d = (key >> shift) & (RADIX - 1);
    uint32_t active = (uint32_t)__ballot(valid);
    uint32_t peers = 0xFFFFFFFFu;
#pragma unroll
    for (int b = 0; b < 8; ++b) {
      uint32_t bit = (d >> b) & 1u;
      uint32_t bal = (uint32_t)__ballot((int)bit);
      peers &= bit ? bal : ~bal;
    }
    peers &= active;
    if (valid) {
      int leader = __builtin_ctz(peers);
      uint32_t cntp = (uint32_t)__popc(peers);
      uint32_t bs = 0;
      if (lane == leader) bs = atomicAdd(&offs[d], cntp);   // ds_add_rtn_u32
      bs = __shfl(bs, leader);
      uint32_t pos = bs + (uint32_t)__popc(peers & ((1u << lane) - 1u));
      keysOut[pos] = key;
      idxOut[pos]  = tI[e];
    }
  }
}

// ---------------------------------------------------------------------------
// Fused tail: gather ids/flags via sorted order, per-group-of-4 selection
// (exact reference semantics), gather output traindata rows.
// tp == (int)neighbor[r,0] (< T always); maxindex==-100 clips to row 0.
// ---------------------------------------------------------------------------
__global__ void k_final(const float* __restrict__ traindata,
                        const float* __restrict__ neighbor,
                        const uint32_t* __restrict__ sortedIdx,
                        float* __restrict__ out, int N, int SEL, int G) {
  int gidx = blockIdx.x * blockDim.x + threadIdx.x;
  if (gidx >= G) return;
  // base element = (N - SEL) + 4*gidx ; divisible by 4 -> uint4 load
  const uint4 r4 = ((const uint4*)sortedIdx)[(uint32_t)(N - SEL) / 4u + (uint32_t)gidx];
  uint32_t rr[4] = { r4.x, r4.y, r4.z, r4.w };
  const float4* td4 = (const float4*)traindata;
  float maxmin = NEG_INIT;
  int   maxindex = -100;
#pragma unroll
  for (int j = 0; j < 4; ++j) {
    uint32_t r = rr[j];
    int   id   = (int)neighbor[(size_t)r * 5 + 0];
    float flag = neighbor[(size_t)r * 5 + 4];
    float4 row = td4[id];
    float mx = fmaxf(row.y, fmaxf(row.z, row.w));
    bool re1 = (flag != 0.0f);
    bool gt  = (mx > maxmin);
    if (re1 == gt) { maxmin = mx; maxindex = id; }
  }
  int mi = (maxindex < 0) ? 0 : maxindex;   // clip(maxindex, 0, ...) ; always < T
  ((float4*)out)[gidx] = td4[mi];
}

// ---------------------------------------------------------------------------
extern "C" void kernel_launch(void* const* d_in, const int* in_sizes, int n_in,
                              void* d_out, int out_size, void* d_ws, size_t ws_size,
                              hipStream_t stream) {
  const float* traindata = (const float*)d_in[0];
  const float* neighbor  = (const float*)d_in[1];
  float* out = (float*)d_out;

  const int T   = in_sizes[0] / 4;   (void)T;
  const int N   = in_sizes[1] / 5;                 // 2,000,000
  const int G   = out_size / 4;                    // 262,144 groups
  const int SEL = G * 4;                           // n = 1,048,576
  const int nblk = (N + TILE - 1) / TILE;          // radix tiles
  const int M    = RADIX * nblk;                   // scan length

  // Workspace layout (256B aligned; +4KB pad per buffer so tail-tile async
  // b128 over-reads stay inside d_ws).
  auto align256 = [](size_t x) { return (x + 255) & ~(size_t)255; };
  const size_t bufBytes = align256((size_t)N * 4 + 4096);
  char* p = (char*)d_ws;
  uint32_t* hist  = (uint32_t*)p; p += align256((size_t)M * 4);
  uint32_t* keysA = (uint32_t*)p; p += bufBytes;
  uint32_t* idxA  = (uint32_t*)p; p += bufBytes;
  uint32_t* keysB = (uint32_t*)p; p += bufBytes;
  uint32_t* idxB  = (uint32_t*)p; p += bufBytes;
  (void)ws_size; (void)n_in;

  k_init<<<(N + 255) / 256, 256, 0, stream>>>(neighbor, keysA, idxA, N);

  uint32_t *ki = keysA, *ii = idxA, *ko = keysB, *io = idxB;
  for (int pass = 0; pass < NPASS; ++pass) {
    const int shift = 8 * pass;
    k_hist   <<<nblk, WAVE,   0, stream>>>(ki, hist, nblk, N, shift);
    k_scan   <<<1,    SCAN_T, 0, stream>>>(hist, M);
    k_scatter<<<nblk, WAVE,   0, stream>>>(ki, ii, ko, io, hist, nblk, N, shift);
    uint32_t* t;
    t = ki; ki = ko; ko = t;
    t = ii; ii = io; io = t;
  }
  // After 4 passes, sorted (key,idx) live in keysA/idxA == (ki, ii).
  k_final<<<(G + 255) / 256, 256, 0, stream>>>(traindata, neighbor, ii, out, N, SEL, G);
}